// QuantizedLinear_74388833567052
// MI455X (gfx1250) — compile-verified
//
#include <hip/hip_runtime.h>

// Problem constants (match the reference).
#define IN_F   4096
#define OUT_F  11008
#define TOKENS 64
#define RANK   16
#define LORA_SCALE 1.0f   // ALPHA / RANK = 16 / 16

typedef __attribute__((ext_vector_type(2))) float v2f;
typedef __attribute__((ext_vector_type(4))) float v4f;
typedef __attribute__((ext_vector_type(8))) float v8f;

// ---------------------------------------------------------------------------
// Kernel 1: xa[t][r] = (alpha/rank) * sum_k x[t][k] * lora_A[r][k]
// Tiny [64,16] intermediate; one block per token, 16 K-chunks x 16 ranks.
// ---------------------------------------------------------------------------
__global__ void __launch_bounds__(256)
lora_xa_kernel(const float* __restrict__ x,
               const float* __restrict__ lora_A,
               float* __restrict__ xa) {
    __shared__ float red[16][17];           // [chunk][rank], padded vs bank conflicts
    const int t     = blockIdx.x;           // token
    const int r     = threadIdx.x & 15;     // rank index
    const int chunk = threadIdx.x >> 4;     // 16 chunks of 256 k each

    const float* xp = x      + t * IN_F + chunk * 256;
    const float* ap = lora_A + r * IN_F + chunk * 256;
    float acc = 0.f;
#pragma unroll 4
    for (int k = 0; k < 256; k += 4) {
        v4f xv = *(const v4f*)(xp + k);
        v4f av = *(const v4f*)(ap + k);
        acc += xv.x * av.x + xv.y * av.y + xv.z * av.z + xv.w * av.w;
    }
    red[chunk][r] = acc;
    __syncthreads();
    if (threadIdx.x < 16) {
        float s = 0.f;
#pragma unroll
        for (int c = 0; c < 16; ++c) s += red[c][threadIdx.x];
        xa[t * RANK + threadIdx.x] = s * LORA_SCALE;
    }
}

// ---------------------------------------------------------------------------
// Kernel 2: main fused GEMM.
//   out[t,o] = scale[o] * (x @ W^T)[t,o] + (xa @ B^T)[t,o] + bias[o]
//
// Wave layout: block = 8 waves = 4 N-tiles x 2 M-halves.
//   wave w: N-tile (w & 3) of this block's 64-channel slab,
//           M-half (w >> 2): tokens [0,32) or [32,64), held as TWO 16x16
//           accumulators so each W load feeds 4 WMMAs.
// Each W row is read by exactly 2 waves, both resident on the same WGP, so
// HBM W traffic is ~180 MB (one pass) by construction — the 23.3 TB/s floor.
// ---------------------------------------------------------------------------
__global__ void __launch_bounds__(256)
qlinear_wmma_kernel(const float* __restrict__ x,
                    const float* __restrict__ w,
                    const float* __restrict__ wscale,
                    const float* __restrict__ lora_B,
                    const float* __restrict__ bias,
                    const float* __restrict__ xa,
                    float* __restrict__ out) {
    __shared__ float xa_lds[TOKENS * RANK];   // all 64 tokens' rank-16 vector (4 KB)

    // Stage xa (1024 floats, 4 per thread) — uniform, no divergence.
    for (int i = threadIdx.x; i < TOKENS * RANK; i += 256)
        xa_lds[i] = xa[i];
    __syncthreads();

    const int wave = threadIdx.x >> 5;
    const int lane = threadIdx.x & 31;
    const int n    = lane & 15;        // N within tile (also A-operand M lane)
    const int h    = lane >> 4;        // lane half: K slots {0,1} vs {2,3}
    const int nt   = wave & 3;         // N-tile within block's 64-channel slab
    const int mh   = wave >> 2;        // M-half: tokens [0,32) or [32,64)

    const int o  = blockIdx.x * 64 + nt * 16 + n;   // output channel
    const int m0 = mh * 32;                         // first token of this half

    // b128 per lane at column kb + h*4: (x,y) -> first WMMA, (z,w) -> second, each at K slots {2h,2h+1}.
    const float* wp  = w + (size_t)o             * IN_F + h * 4;
    const float* xp0 = x + (size_t)(m0 + n)      * IN_F + h * 4;   // M-tile 0
    const float* xp1 = x + (size_t)(m0 + 16 + n) * IN_F + h * 4;   // M-tile 1

    v8f c0 = {}, c1 = {};   // two 16x16 f32 accumulators
#pragma unroll 2
    for (int kb = 0; kb < IN_F; kb += 8) {
        v4f bv  = *(const v4f*)(wp  + kb);
        v4f a0v = *(const v4f*)(xp0 + kb);
        v4f a1v = *(const v4f*)(xp1 + kb);
        v2f blo = {bv.x,  bv.y},  bhi = {bv.z,  bv.w};
        v2f a0l = {a0v.x, a0v.y}, a0h = {a0v.z, a0v.w};
        v2f a1l = {a1v.x, a1v.y}, a1h = {a1v.z, a1v.w};
        // Interleave the two independent accumulation chains.
        c0 = __builtin_amdgcn_wmma_f32_16x16x4_f32(false, a0l, false, blo,
                                                   (short)0, c0, false, false);
        c1 = __builtin_amdgcn_wmma_f32_16x16x4_f32(false, a1l, false, blo,
                                                   (short)0, c1, false, false);
        c0 = __builtin_amdgcn_wmma_f32_16x16x4_f32(false, a0h, false, bhi,
                                                   (short)0, c0, false, false);
        c1 = __builtin_amdgcn_wmma_f32_16x16x4_f32(false, a1h, false, bhi,
                                                   (short)0, c1, false, false);
    }

    // --- Epilogue: per-channel scale, LoRA rank-16 dot, bias ---
    const float sc = wscale[o];
    const float bs = bias[o];
    const v4f* brow4 = (const v4f*)(lora_B + (size_t)o * RANK);   // 64B-aligned
    v4f bA = brow4[0], bB = brow4[1], bC = brow4[2], bD = brow4[3];
    float br[16] = {bA.x, bA.y, bA.z, bA.w, bB.x, bB.y, bB.z, bB.w,
                    bC.x, bC.y, bC.z, bC.w, bD.x, bD.y, bD.z, bD.w};

    // C layout: VGPR r -> token row (base + r + 8*h), column n.
#pragma unroll
    for (int tile = 0; tile < 2; ++tile) {
        const v8f cc = tile ? c1 : c0;
        const int tbase = m0 + tile * 16 + h * 8;
        float* op = out + (size_t)tbase * OUT_F + o;
#pragma unroll
        for (int r = 0; r < 8; ++r) {
            const float* xr = &xa_lds[(tbase + r) * RANK];
            float lora = 0.f;
#pragma unroll
            for (int j = 0; j < 16; ++j) lora += xr[j] * br[j];
            op[(size_t)r * OUT_F] = cc[r] * sc + lora + bs;
        }
    }
}

// ---------------------------------------------------------------------------
extern "C" void kernel_launch(void* const* d_in, const int* in_sizes, int n_in,
                              void* d_out, int out_size, void* d_ws, size_t ws_size,
                              hipStream_t stream) {
    const float* x       = (const float*)d_in[0];   // [64, 4096]
    const float* w       = (const float*)d_in[1];   // [11008, 4096]
    const float* wscale  = (const float*)d_in[2];   // [11008]
    const float* lora_A  = (const float*)d_in[3];   // [16, 4096]
    const float* lora_B  = (const float*)d_in[4];   // [11008, 16]
    const float* bias    = (const float*)d_in[5];   // [11008]
    float*       out     = (float*)d_out;           // [64, 11008]
    float*       xa      = (float*)d_ws;            // [64, 16] scratch

    lora_xa_kernel<<<TOKENS, 256, 0, stream>>>(x, lora_A, xa);

    qlinear_wmma_kernel<<<OUT_F / 64, 256, 0, stream>>>(x, w, wscale, lora_B,
                                                        bias, xa, out);
}